// HilbertScan3DMambaBlock_5403068858467
// MI455X (gfx1250) — compile-verified
//
#include <hip/hip_runtime.h>
#include <hip/hip_bf16.h>

// ---- problem constants (match reference) ----
#define BB    2
#define NFR   8
#define LL    8192           // NF*H*W
#define DIMC  384
#define DI    192
#define DS    16
#define DCONV 4
#define DTR   24
#define HID   1536
#define TT    (BB*LL)        // 16384 tokens
#define T2    (2*BB*LL)      // 32768 tokens (fwd + reversed stacked)
#define DBLP  64             // padded row stride for dbl (valid cols = 56)
#define NCH   32             // chunks per sequence for the parallel scan
#define SCH   256            // chunk length (NCH*SCH == LL)

typedef float v2f  __attribute__((ext_vector_type(2)));
typedef float v8f  __attribute__((ext_vector_type(8)));
typedef unsigned int u32x4 __attribute__((ext_vector_type(4)));
typedef int   i32x4 __attribute__((ext_vector_type(4)));
typedef int   i32x8 __attribute__((ext_vector_type(8)));

// Tensor Data Mover availability (compile-safe guards; cooperative fallback).
#if defined(__HIP_DEVICE_COMPILE__) && __has_builtin(__builtin_amdgcn_tensor_load_to_lds) && __has_builtin(__builtin_amdgcn_s_wait_tensorcnt)
#define USE_TDM 1
#else
#define USE_TDM 0
#endif
#if __has_include(<hip/amd_detail/amd_gfx1250_TDM.h>)
#define TDM_6ARG 1            // therock-10.0 headers -> 6-arg builtin
#else
#define TDM_6ARG 0            // ROCm 7.2 -> 5-arg builtin
#endif

__device__ __forceinline__ float siluf(float x) { return x / (1.f + __expf(-x)); }

// ---------------------------------------------------------------------------
// Stage one chunk's B/C slab (SCH rows x 32 floats = 32KB) from dbl into LDS.
// TDM path: one DMA descriptor per block (issued by wave 0, EXEC-independent),
// D# per CDNA5 ISA 8.3/8.4: 2-D tile, data_size=4B, tile 32x256, stride 64.
// ---------------------------------------------------------------------------
__device__ __forceinline__ void stage_bc(const float* __restrict__ dbl,
                                         size_t row0, float* sBC)
{
#if USE_TDM
    if (threadIdx.x < 32) {   // one wave issues the tensor DMA
        unsigned long long ga =
            (unsigned long long)(uintptr_t)(dbl + row0 * DBLP + DTR);
        unsigned lds = (unsigned)(uintptr_t)sBC;   // low 32 bits = LDS byte addr
        u32x4 g0 = { 1u,                                  // count=1, user D#
                     lds,                                 // lds_addr
                     (unsigned)(ga & 0xffffffffu),        // global_addr[31:0]
                     (unsigned)((ga >> 32) & 0x1ffffffu)  // global_addr[56:32]
                       | (2u << 30) };                    // type=2 ("image")
        i32x8 g1 = { (int)(2u << 16),        // data_size = 2 (4 bytes)
                     (int)(32u << 16),       // tensor_dim0 = 32 (bits 79:48 lo)
                     (int)(SCH << 16),       // tensor_dim1 = 256 (bits 95:80)
                     (int)(32u << 16),       // tile_dim0 = 32 (bits 127:112)
                     (int)SCH,               // tile_dim1 = 256 (bits 143:128)
                     (int)DBLP,              // tensor_dim0_stride = 64
                     0, 0 };
        i32x4 gz = { 0, 0, 0, 0 };           // <=2D: groups 2/3 disabled
#if TDM_6ARG
        i32x8 gz8 = { 0, 0, 0, 0, 0, 0, 0, 0 };
        __builtin_amdgcn_tensor_load_to_lds(g0, g1, gz, gz, gz8, 0);
#else
        __builtin_amdgcn_tensor_load_to_lds(g0, g1, gz, gz, 0);
#endif
        __builtin_amdgcn_s_wait_tensorcnt(0);
    }
    __syncthreads();
#else
    for (int i = threadIdx.x; i < SCH * 32; i += 64) {
        int r = i >> 5, k = i & 31;
        sBC[i] = dbl[(row0 + r) * DBLP + DTR + k];
    }
    __syncthreads();
#endif
}

// ---------------------------------------------------------------------------
// K1: hilbert gather + LayerNorm1; also emit flipped copy for reverse mamba
// ---------------------------------------------------------------------------
__global__ void k_gather_ln(const float* __restrict__ x, const float* __restrict__ fx,
                            const int* __restrict__ hc,
                            const float* __restrict__ w, const float* __restrict__ bi,
                            float* __restrict__ x_flat, float* __restrict__ f_flat,
                            float* __restrict__ normedAll)
{
    __shared__ float red[128];
    int t = blockIdx.x;            // token 0..TT-1
    int b = t / LL, l = t % LL;
    int p = hc[l];
    int f = p >> 10, hw = p & 1023;
    size_t base = ((size_t)(b * NFR + f) * DIMC) * 1024 + hw;
    int tid = threadIdx.x;

    float v0[3];
    float s = 0.f;
    #pragma unroll
    for (int i = 0; i < 3; ++i) {
        int d = tid + i * 128;
        float v = x[base + (size_t)d * 1024];
        v0[i] = v; s += v;
        x_flat[(size_t)t * DIMC + d] = v;
        f_flat[(size_t)t * DIMC + d] = fx[base + (size_t)d * 1024];
    }
    red[tid] = s; __syncthreads();
    for (int off = 64; off > 0; off >>= 1) { if (tid < off) red[tid] += red[tid + off]; __syncthreads(); }
    float mean = red[0] * (1.f / DIMC);
    __syncthreads();
    s = 0.f;
    #pragma unroll
    for (int i = 0; i < 3; ++i) { float dv = v0[i] - mean; s += dv * dv; }
    red[tid] = s; __syncthreads();
    for (int off = 64; off > 0; off >>= 1) { if (tid < off) red[tid] += red[tid + off]; __syncthreads(); }
    float rstd = rsqrtf(red[0] * (1.f / DIMC) + 1e-5f);

    size_t rf = (size_t)t * DIMC;
    size_t rr = ((size_t)(2 + b) * LL + (LL - 1 - l)) * DIMC;
    #pragma unroll
    for (int i = 0; i < 3; ++i) {
        int d = tid + i * 128;
        float nv = (v0[i] - mean) * rstd * w[d] + bi[d];
        normedAll[rf + d] = nv;
        normedAll[rr + d] = nv;
    }
}

// ---------------------------------------------------------------------------
// Generic row LayerNorm (dim = 384)
// ---------------------------------------------------------------------------
__global__ void k_layernorm(const float* __restrict__ in, const float* __restrict__ w,
                            const float* __restrict__ bi, float* __restrict__ out)
{
    __shared__ float red[128];
    size_t row = (size_t)blockIdx.x * DIMC;
    int tid = threadIdx.x;
    float v0[3];
    float s = 0.f;
    #pragma unroll
    for (int i = 0; i < 3; ++i) { float v = in[row + tid + i * 128]; v0[i] = v; s += v; }
    red[tid] = s; __syncthreads();
    for (int off = 64; off > 0; off >>= 1) { if (tid < off) red[tid] += red[tid + off]; __syncthreads(); }
    float mean = red[0] * (1.f / DIMC);
    __syncthreads();
    s = 0.f;
    #pragma unroll
    for (int i = 0; i < 3; ++i) { float dv = v0[i] - mean; s += dv * dv; }
    red[tid] = s; __syncthreads();
    for (int off = 64; off > 0; off >>= 1) { if (tid < off) red[tid] += red[tid + off]; __syncthreads(); }
    float rstd = rsqrtf(red[0] * (1.f / DIMC) + 1e-5f);
    #pragma unroll
    for (int i = 0; i < 3; ++i) {
        int d = tid + i * 128;
        out[row + d] = (v0[i] - mean) * rstd * w[d] + bi[d];
    }
}

// ---------------------------------------------------------------------------
// Generic fp32 WMMA GEMM:  C[M,N] = A[M,K] @ W[N,K]^T (+bias) (+addsrc) (+act)
// One wave computes a 16x64 strip (4 accumulators), reusing the A fragment
// across 4 v_wmma_f32_16x16x4_f32 per k-step. 4 waves/block cover 64 rows.
// Masked (not branched) B loads keep EXEC all-ones around the WMMA ops.
// act: 0 = none, 1 = softplus
// ---------------------------------------------------------------------------
__global__ void k_gemm_wmma(const float* __restrict__ A, int lda,
                            const float* __restrict__ W, int K,
                            const float* __restrict__ bias,
                            const float* __restrict__ addsrc,
                            float* __restrict__ C, int ldc,
                            int Nvalid, int act)
{
    int lane = threadIdx.x & 31;
    int wave = threadIdx.x >> 5;
    int half = lane >> 4;
    int lm   = lane & 15;
    int n0 = blockIdx.x * 64;
    int m0 = (blockIdx.y * 4 + wave) * 16;

    const float* ap = A + (size_t)(m0 + lm) * lda + 2 * half;
    const float* wp[4];
    float msk[4];
    #pragma unroll
    for (int t = 0; t < 4; ++t) {
        int col  = n0 + t * 16 + lm;
        int colc = (col < Nvalid) ? col : (Nvalid - 1);
        msk[t]   = (col < Nvalid) ? 1.f : 0.f;
        wp[t]    = W + (size_t)colc * K + 2 * half;
    }

    v8f zz = {0.f, 0.f, 0.f, 0.f, 0.f, 0.f, 0.f, 0.f};
    v8f acc[4];
    #pragma unroll
    for (int t = 0; t < 4; ++t) acc[t] = zz;

    for (int k0 = 0; k0 < K; k0 += 4) {
        v2f a;
        a[0] = ap[k0];
        a[1] = ap[k0 + 1];
        #pragma unroll
        for (int t = 0; t < 4; ++t) {
            v2f bfr;
            bfr[0] = wp[t][k0]     * msk[t];
            bfr[1] = wp[t][k0 + 1] * msk[t];
            acc[t] = __builtin_amdgcn_wmma_f32_16x16x4_f32(
                false, a, false, bfr, (short)0, acc[t], false, false);
        }
    }

    #pragma unroll
    for (int t = 0; t < 4; ++t) {
        int col = n0 + t * 16 + lm;
        if (col < Nvalid) {
            float bv = bias ? bias[col] : 0.f;
            #pragma unroll
            for (int j = 0; j < 8; ++j) {
                int row = m0 + j + 8 * half;   // C/D layout: VGPR j -> M=j / M=j+8
                float v = acc[t][j] + bv;
                if (addsrc) v += addsrc[(size_t)row * ldc + col];
                if (act == 1) v = (v > 20.f) ? v : log1pf(__expf(v));
                C[(size_t)row * ldc + col] = v;
            }
        }
    }
}

// ---------------------------------------------------------------------------
// K3: causal depthwise conv1d (k=4, left pad 3) + SiLU, xp = xz[..., :192]
// ---------------------------------------------------------------------------
__global__ void k_conv1d_silu(const float* __restrict__ xz, const float* __restrict__ cw,
                              const float* __restrict__ cb, float* __restrict__ xc)
{
    int idx = blockIdx.x * blockDim.x + threadIdx.x;
    if (idx >= T2 * DI) return;
    int c = idx % DI;
    int g = idx / DI;     // global token across 4 stacked sequences
    int l = g % LL;
    float acc = cb[c];
    #pragma unroll
    for (int j = 0; j < DCONV; ++j) {
        int li = l - (DCONV - 1) + j;
        if (li >= 0) acc += cw[c * DCONV + j] * xz[(size_t)(g - l + li) * (2 * DI) + c];
    }
    xc[idx] = siluf(acc);
}

// ---------------------------------------------------------------------------
// Chunked parallel selective scan. h <- exp(dt*A)*h + dt*B*x is affine in h,
// so: pass1 composes each 256-step chunk (Sigma dt  and  b_comp[16]);
//     mid scans the 32 chunk summaries to get each chunk's initial state;
//     pass2 replays chunks in parallel emitting y = <h, C>.
// Sequential latency: 8192 -> ~2*256+32 steps.
// ---------------------------------------------------------------------------
__global__ void k_scan_pass1(const float* __restrict__ dbl, const float* __restrict__ dt,
                             const float* __restrict__ xc, const float* __restrict__ A_log,
                             float* __restrict__ sumdt, float* __restrict__ bcomp)
{
    __shared__ float sBC[SCH * 32];
    int blk = blockIdx.x;              // s-major: s*(NCH*3) + c*3 + g
    int s = blk / (NCH * 3);
    int rem = blk % (NCH * 3);
    int c = rem / 3;
    int g = rem % 3;
    int d = g * 64 + threadIdx.x;
    size_t row0 = (size_t)s * LL + (size_t)c * SCH;

    stage_bc(dbl, row0, sBC);

    float Ar[DS], bacc[DS];
    #pragma unroll
    for (int j = 0; j < DS; ++j) { Ar[j] = -__expf(A_log[d * DS + j]); bacc[j] = 0.f; }

    float sdt = 0.f;
    for (int ll = 0; ll < SCH; ++ll) {
        size_t base = row0 + ll;
        float dtv = dt[base * DI + d];
        float xv  = xc[base * DI + d];
        sdt += dtv;
        const float* Bp = &sBC[ll * 32];
        #pragma unroll
        for (int j = 0; j < DS; ++j) {
            float a = __expf(dtv * Ar[j]);
            bacc[j] = a * bacc[j] + (dtv * Bp[j]) * xv;
        }
    }
    size_t o = ((size_t)(s * NCH + c) * DI + d) * DS;
    #pragma unroll
    for (int j = 0; j < DS; ++j) bcomp[o + j] = bacc[j];
    sumdt[(s * NCH + c) * DI + d] = sdt;
}

__global__ void k_scan_mid(const float* __restrict__ sumdt, const float* __restrict__ bcomp,
                           const float* __restrict__ A_log, float* __restrict__ hinit)
{
    int idx = blockIdx.x * 64 + threadIdx.x;   // 0..767 : (s, d)
    int s = idx / DI;
    int d = idx % DI;
    float Ar[DS], h[DS];
    #pragma unroll
    for (int j = 0; j < DS; ++j) { Ar[j] = -__expf(A_log[d * DS + j]); h[j] = 0.f; }
    for (int c = 0; c < NCH; ++c) {
        size_t o = ((size_t)(s * NCH + c) * DI + d) * DS;
        float sd = sumdt[(s * NCH + c) * DI + d];
        #pragma unroll
        for (int j = 0; j < DS; ++j) {
            hinit[o + j] = h[j];
            h[j] = __expf(sd * Ar[j]) * h[j] + bcomp[o + j];
        }
    }
}

__global__ void k_scan_pass2(const float* __restrict__ dbl, const float* __restrict__ dt,
                             const float* __restrict__ xc, const float* __restrict__ A_log,
                             const float* __restrict__ hinit, float* __restrict__ ys)
{
    __shared__ float sBC[SCH * 32];
    int blk = blockIdx.x;
    int s = blk / (NCH * 3);
    int rem = blk % (NCH * 3);
    int c = rem / 3;
    int g = rem % 3;
    int d = g * 64 + threadIdx.x;
    size_t row0 = (size_t)s * LL + (size_t)c * SCH;

    stage_bc(dbl, row0, sBC);

    size_t o = ((size_t)(s * NCH + c) * DI + d) * DS;
    float Ar[DS], h[DS];
    #pragma unroll
    for (int j = 0; j < DS; ++j) { Ar[j] = -__expf(A_log[d * DS + j]); h[j] = hinit[o + j]; }

    for (int ll = 0; ll < SCH; ++ll) {
        size_t base = row0 + ll;
        float dtv = dt[base * DI + d];
        float xv  = xc[base * DI + d];
        const float* Bp = &sBC[ll * 32];
        float y = 0.f;
        #pragma unroll
        for (int j = 0; j < DS; ++j) {
            float a = __expf(dtv * Ar[j]);
            h[j] = a * h[j] + (dtv * Bp[j]) * xv;
            y += h[j] * Bp[16 + j];
        }
        ys[base * DI + d] = y;
    }
}

// ---------------------------------------------------------------------------
// K5: y = (y + Dp*xc) * silu(z),  z = xz[..., 192:]
// ---------------------------------------------------------------------------
__global__ void k_gate(float* __restrict__ ys, const float* __restrict__ xc,
                       const float* __restrict__ xz, const float* __restrict__ Dp)
{
    int idx = blockIdx.x * blockDim.x + threadIdx.x;
    if (idx >= T2 * DI) return;
    int c = idx % DI;
    int g = idx / DI;
    float z = xz[(size_t)g * (2 * DI) + DI + c];
    ys[idx] = (ys[idx] + Dp[c] * xc[idx]) * siluf(z);
}

// ---------------------------------------------------------------------------
// K7a: pooled[b,s,d] = mean_l seq_s[b,l,d]   (mean is flip-invariant)
// ---------------------------------------------------------------------------
__global__ void k_pool(const float* __restrict__ m_all, const float* __restrict__ f_flat,
                       float* __restrict__ pooled)
{
    __shared__ float red[256];
    int o = blockIdx.x;                   // b*1152 + sq*384 + d
    int d  = o % DIMC;
    int sq = (o / DIMC) % 3;
    int b  = o / (DIMC * 3);
    const float* src = (sq == 2) ? f_flat : m_all;
    size_t row0 = (sq == 1) ? (size_t)(2 + b) * LL : (size_t)b * LL;
    float s = 0.f;
    for (int l = threadIdx.x; l < LL; l += 256) s += src[(row0 + l) * DIMC + d];
    red[threadIdx.x] = s; __syncthreads();
    for (int off = 128; off > 0; off >>= 1) {
        if (threadIdx.x < off) red[threadIdx.x] += red[threadIdx.x + off];
        __syncthreads();
    }
    if (threadIdx.x == 0) pooled[o] = red[0] * (1.f / LL);
}

// K7b: softmax over the 3 branch logits per (b,d)
__global__ void k_ssa_weights(const float* __restrict__ pooled, const float* __restrict__ ssa_w,
                              float* __restrict__ aw)
{
    int b = blockIdx.x;
    int d = threadIdx.x;      // 384
    float lg[3];
    #pragma unroll
    for (int t = 0; t < 3; ++t) {
        float s = 0.f;
        #pragma unroll
        for (int sq = 0; sq < 3; ++sq)
            s += pooled[(b * 3 + sq) * DIMC + d] * ssa_w[d * 9 + t * 3 + sq];
        lg[t] = s;
    }
    float mx = fmaxf(lg[0], fmaxf(lg[1], lg[2]));
    float e0 = __expf(lg[0] - mx), e1 = __expf(lg[1] - mx), e2 = __expf(lg[2] - mx);
    float inv = 1.f / (e0 + e1 + e2);
    aw[(b * 3 + 0) * DIMC + d] = e0 * inv;
    aw[(b * 3 + 1) * DIMC + d] = e1 * inv;
    aw[(b * 3 + 2) * DIMC + d] = e2 * inv;
}

// K7c: xm = x_flat + sum_t aw[t]*branch_t   (branch 1 read time-flipped)
__global__ void k_combine(const float* __restrict__ m_all, const float* __restrict__ f_flat,
                          const float* __restrict__ x_flat, const float* __restrict__ aw,
                          float* __restrict__ xm)
{
    int idx = blockIdx.x * blockDim.x + threadIdx.x;
    if (idx >= TT * DIMC) return;
    int d = idx % DIMC;
    int t = idx / DIMC;
    int b = t / LL, l = t % LL;
    float mf = m_all[(size_t)t * DIMC + d];
    float mr = m_all[((size_t)(2 + b) * LL + (LL - 1 - l)) * DIMC + d];
    float ff = f_flat[idx];
    float mo = aw[(b * 3 + 0) * DIMC + d] * mf
             + aw[(b * 3 + 1) * DIMC + d] * mr
             + aw[(b * 3 + 2) * DIMC + d] * ff;
    xm[idx] = x_flat[idx] + mo;
}

// ---------------------------------------------------------------------------
// K10: 3x3x3 depthwise conv over (NF,H,W) in hilbert-scan layout + exact GELU
// ---------------------------------------------------------------------------
__global__ void k_dwconv_gelu(const float* __restrict__ h1, const float* __restrict__ dw,
                              const float* __restrict__ db, float* __restrict__ h1c)
{
    size_t idx = (size_t)blockIdx.x * blockDim.x + threadIdx.x;
    if (idx >= (size_t)TT * HID) return;
    int e = (int)(idx % HID);
    int t = (int)(idx / HID);
    int b = t / LL, l = t % LL;
    int f = l >> 10, h = (l >> 5) & 31, wv = l & 31;
    // speculative prefetch of the next frame plane (global_prefetch_b8)
    __builtin_prefetch(h1 + ((size_t)b * LL + l + 1024) * HID + e, 0, 0);
    float acc = db[e];
    #pragma unroll
    for (int dz = -1; dz <= 1; ++dz) {
        int f2 = f + dz; if (f2 < 0 || f2 >= NFR) continue;
        #pragma unroll
        for (int dy = -1; dy <= 1; ++dy) {
            int h2 = h + dy; if ((unsigned)h2 >= 32u) continue;
            #pragma unroll
            for (int dx = -1; dx <= 1; ++dx) {
                int w2 = wv + dx; if ((unsigned)w2 >= 32u) continue;
                int l2 = (f2 << 10) + (h2 << 5) + w2;
                acc += dw[e * 27 + (dz + 1) * 9 + (dy + 1) * 3 + (dx + 1)]
                     * h1[((size_t)b * LL + l2) * HID + e];
            }
        }
    }
    h1c[idx] = 0.5f * acc * (1.f + erff(acc * 0.70710678118f));
}

// ---------------------------------------------------------------------------
// K12: inverse hilbert scatter back to (B, NF, DIM, H, W)
// ---------------------------------------------------------------------------
__global__ void k_scatter(const float* __restrict__ fin, const int* __restrict__ hc,
                          float* __restrict__ out)
{
    int idx = blockIdx.x * blockDim.x + threadIdx.x;
    if (idx >= TT * DIMC) return;
    int d = idx % DIMC;
    int t = idx / DIMC;
    int b = t / LL, l = t % LL;
    int p = hc[l];
    int f = p >> 10, hw = p & 1023;
    out[((size_t)(b * NFR + f) * DIMC + d) * 1024 + hw] = fin[idx];
}

// ---------------------------------------------------------------------------
extern "C" void kernel_launch(void* const* d_in, const int* in_sizes, int n_in,
                              void* d_out, int out_size, void* d_ws, size_t ws_size,
                              hipStream_t stream)
{
    const float* x        = (const float*)d_in[0];
    const float* freq_x   = (const float*)d_in[1];
    const int*   hc       = (const int*)  d_in[2];
    const float* ln1_w    = (const float*)d_in[3];
    const float* ln1_b    = (const float*)d_in[4];
    const float* in_proj  = (const float*)d_in[5];   // (384, 384)
    const float* conv_w   = (const float*)d_in[6];   // (192, 4)
    const float* conv_b   = (const float*)d_in[7];
    const float* xproj_w  = (const float*)d_in[8];   // (56, 192)
    const float* dt_w     = (const float*)d_in[9];   // (192, 24)
    const float* dt_b     = (const float*)d_in[10];
    const float* A_log    = (const float*)d_in[11];  // (192, 16)
    const float* Dp       = (const float*)d_in[12];
    const float* out_proj = (const float*)d_in[13];  // (384, 192)
    const float* ssa_w    = (const float*)d_in[14];  // (384, 3, 3)
    const float* ln2_w    = (const float*)d_in[15];
    const float* ln2_b    = (const float*)d_in[16];
    const float* fc1_w    = (const float*)d_in[17];  // (1536, 384)
    const float* fc1_b    = (const float*)d_in[18];
    const float* dw_w     = (const float*)d_in[19];  // (1536,1,3,3,3)
    const float* dw_b     = (const float*)d_in[20];
    const float* fc2_w    = (const float*)d_in[21];  // (384, 1536)
    const float* fc2_b    = (const float*)d_in[22];
    float* out = (float*)d_out;
    (void)in_sizes; (void)n_in; (void)out_size; (void)ws_size;

    float* ws = (float*)d_ws;
    size_t off = 0;
    auto carve = [&](size_t n) { float* p = ws + off; off += n; return p; };
    float* x_flat    = carve((size_t)TT * DIMC);
    float* f_flat    = carve((size_t)TT * DIMC);
    float* normedAll = carve((size_t)T2 * DIMC);
    float* xz        = carve((size_t)T2 * 2 * DI);
    float* xc        = carve((size_t)T2 * DI);
    float* dbl       = carve((size_t)T2 * DBLP);
    float* dtb       = carve((size_t)T2 * DI);
    float* ys        = carve((size_t)T2 * DI);
    float* m_all     = carve((size_t)T2 * DIMC);
    float* pooled    = carve((size_t)2 * 3 * DIMC);
    float* aw        = carve((size_t)2 * 3 * DIMC);
    float* xm        = carve((size_t)TT * DIMC);
    float* ln2n      = carve((size_t)TT * DIMC);
    float* h1        = carve((size_t)TT * HID);
    float* h1c       = carve((size_t)TT * HID);
    float* fin       = carve((size_t)TT * DIMC);
    float* sumdt     = carve((size_t)4 * NCH * DI);
    float* bcomp     = carve((size_t)4 * NCH * DI * DS);
    float* hinit     = carve((size_t)4 * NCH * DI * DS);

    // 1) gather + LN1 (+ flipped copy)
    k_gather_ln<<<TT, 128, 0, stream>>>(x, freq_x, hc, ln1_w, ln1_b,
                                        x_flat, f_flat, normedAll);

    // 2) xz = normedAll @ in_proj^T   (32768 x 384, K=384)
    k_gemm_wmma<<<dim3(DIMC / 64, T2 / 64), 128, 0, stream>>>(
        normedAll, DIMC, in_proj, DIMC, nullptr, nullptr, xz, 2 * DI, 2 * DI, 0);

    // 3) causal conv1d + SiLU
    k_conv1d_silu<<<(T2 * DI + 255) / 256, 256, 0, stream>>>(xz, conv_w, conv_b, xc);

    // 4) dbl = xc @ xproj^T  (N=56 padded to 64)
    k_gemm_wmma<<<dim3(1, T2 / 64), 128, 0, stream>>>(
        xc, DI, xproj_w, DI, nullptr, nullptr, dbl, DBLP, DTR + 2 * DS, 0);

    // 5) dt = softplus(dbl[:, :24] @ dt_w^T + dt_b)
    k_gemm_wmma<<<dim3(DI / 64, T2 / 64), 128, 0, stream>>>(
        dbl, DBLP, dt_w, DTR, dt_b, nullptr, dtb, DI, DI, 1);

    // 6) chunked parallel selective scan (TDM-staged B/C slabs)
    k_scan_pass1<<<4 * NCH * 3, 64, 0, stream>>>(dbl, dtb, xc, A_log, sumdt, bcomp);
    k_scan_mid<<<12, 64, 0, stream>>>(sumdt, bcomp, A_log, hinit);
    k_scan_pass2<<<4 * NCH * 3, 64, 0, stream>>>(dbl, dtb, xc, A_log, hinit, ys);

    // 7) gate: (y + Dp*xc) * silu(z)
    k_gate<<<(T2 * DI + 255) / 256, 256, 0, stream>>>(ys, xc, xz, Dp);

    // 8) m_all = y @ out_proj^T  (K=192)
    k_gemm_wmma<<<dim3(DIMC / 64, T2 / 64), 128, 0, stream>>>(
        ys, DI, out_proj, DI, nullptr, nullptr, m_all, DIMC, DIMC, 0);

    // 9) SSA
    k_pool<<<2 * 3 * DIMC, 256, 0, stream>>>(m_all, f_flat, pooled);
    k_ssa_weights<<<2, DIMC, 0, stream>>>(pooled, ssa_w, aw);
    k_combine<<<(TT * DIMC + 255) / 256, 256, 0, stream>>>(m_all, f_flat, x_flat, aw, xm);

    // 10) LN2
    k_layernorm<<<TT, 128, 0, stream>>>(xm, ln2_w, ln2_b, ln2n);

    // 11) h1 = ln2n @ fc1^T + fc1_b  (16384 x 1536, K=384)
    k_gemm_wmma<<<dim3(HID / 64, TT / 64), 128, 0, stream>>>(
        ln2n, DIMC, fc1_w, DIMC, fc1_b, nullptr, h1, HID, HID, 0);

    // 12) depthwise 3x3x3 conv + GELU
    k_dwconv_gelu<<<(int)(((size_t)TT * HID + 255) / 256), 256, 0, stream>>>(h1, dw_w, dw_b, h1c);

    // 13) fin = h1c @ fc2^T + fc2_b + xm  (K=1536)
    k_gemm_wmma<<<dim3(DIMC / 64, TT / 64), 128, 0, stream>>>(
        h1c, HID, fc2_w, HID, fc2_b, xm, fin, DIMC, DIMC, 0);

    // 14) inverse hilbert scatter to output layout
    k_scatter<<<(TT * DIMC + 255) / 256, 256, 0, stream>>>(fin, hc, out);
}